// QuantumCircuit_11931419149169
// MI455X (gfx1250) — compile-verified
//
#include <hip/hip_runtime.h>

// ---------------------------------------------------------------------------
// Quantum circuit (24 qubits, depth 4, 3 entangling layers) on MI455X/gfx1250.
//
// Algebraic reduction: diagonals factor per qubit, entanglers act on disjoint
// (2p,2p+1) pairs and commute => final = (⊗_{p=0..11} A_p) state0 with A_p a
// 4x4 complex matrix. Three LDS-tiled passes over the 128MB state:
//   pass1: pairs 0..5  (bits 0..11,  contiguous 4096-amplitude tiles)
//   pass2: pairs 6..8  (bits 12..17, 64x64 tiles, rows strided 2^12)
//   pass3: pairs 9..11 (bits 18..23, 64x64 tiles, rows strided 2^18)
//          + fused reductions (norm, entropy, z, overlap) + probs[:384]
// ALL pair applications run on v_wmma_f32_16x16x4_f32 (8x8 real embedding of
// the 4x4 complex matrix, 16-group batches); all tile movement uses gfx1250
// async global<->LDS ops (ASYNCcnt). Workspace: 1MB header + 2*64MB planes.
// ---------------------------------------------------------------------------

#define DIM (1u << 24)

typedef float v2f __attribute__((ext_vector_type(2)));
typedef float v8f __attribute__((ext_vector_type(8)));

#if __has_builtin(__builtin_amdgcn_wmma_f32_16x16x4_f32)
#define USE_WMMA 1
#else
#define USE_WMMA 0
#endif

// ----------------------------- gfx1250 helpers -----------------------------

__device__ __forceinline__ unsigned lds_addr(const void* p) {
  // generic pointer to LDS: low 32 bits are the LDS byte offset
  return (unsigned)(unsigned long long)p;
}
__device__ __forceinline__ void async_ld_b128(unsigned lds, const void* g) {
  asm volatile("global_load_async_to_lds_b128 %0, %1, off"
               :: "v"(lds), "v"((unsigned long long)g) : "memory");
}
__device__ __forceinline__ void async_ld_b64(unsigned lds, const void* g) {
  asm volatile("global_load_async_to_lds_b64 %0, %1, off"
               :: "v"(lds), "v"((unsigned long long)g) : "memory");
}
__device__ __forceinline__ void async_st_b128(void* g, unsigned lds) {
  asm volatile("global_store_async_from_lds_b128 %0, %1, off"
               :: "v"((unsigned long long)g), "v"(lds) : "memory");
}
__device__ __forceinline__ void async_st_b64(void* g, unsigned lds) {
  asm volatile("global_store_async_from_lds_b64 %0, %1, off"
               :: "v"((unsigned long long)g), "v"(lds) : "memory");
}
__device__ __forceinline__ void wait_async0() {
  asm volatile("s_wait_asynccnt 0" ::: "memory");
}

// ------------------------------- prep kernel -------------------------------
// Thread p (0..11) builds A_p = D3*E2*D2*E1*D1*E0*D0 (4x4 complex), stores
// it densely (mats) and pre-swizzled into the v_wmma_f32_16x16x4_f32
// B-operand lane layout (B = M^T with M the 8x8 real embedding of A_p).

__global__ void prep_kernel(const float* __restrict__ rot,
                            const float* __restrict__ entw,
                            float* __restrict__ mats,
                            float* __restrict__ bfrag) {
  int p = threadIdx.x;
  if (p >= 12) return;
  float Ar[4][4] = {}, Ai[4][4] = {};
  Ar[0][0] = Ar[1][1] = Ar[2][2] = Ar[3][3] = 1.f;
  for (int l = 0; l < 4; ++l) {
    float fr[2], fi[2];
    for (int t = 0; t < 2; ++t) {
      const float* rp = rot + ((l * 24) + (2 * p + t)) * 3;
      float h0 = rp[0] * 0.5f, h1 = rp[1] * 0.5f, h2 = rp[2] * 0.5f;
      fr[t] = cosf(h0) * cosf(h1) * cosf(h2);
      fi[t] = sinf(h0) * sinf(h1) * sinf(h2);
    }
    // diagonal: d[m] = g_ctrl(m&1) * g_tgt(m>>1), g(b) = b ? f : conj(f)
    for (int m = 0; m < 4; ++m) {
      float g0r = fr[0], g0i = (m & 1) ? fi[0] : -fi[0];
      float g1r = fr[1], g1i = (m & 2) ? fi[1] : -fi[1];
      float dr = g0r * g1r - g0i * g1i;
      float di = g0r * g1i + g0i * g1r;
      for (int c = 0; c < 4; ++c) {
        float ar = Ar[m][c], ai = Ai[m][c];
        Ar[m][c] = dr * ar - di * ai;
        Ai[m][c] = dr * ai + di * ar;
      }
    }
    if (l < 3) {  // entangler on ctrl==1 rows (m=1 and m=3), beta = i*sin(w)
      float w = entw[l * 12 + p];
      float al = cosf(w), be = sinf(w);
      for (int c = 0; c < 4; ++c) {
        float r1r = Ar[1][c], r1i = Ai[1][c];
        float r3r = Ar[3][c], r3i = Ai[3][c];
        Ar[1][c] = al * r1r - be * r3i;
        Ai[1][c] = al * r1i + be * r3r;
        Ar[3][c] = al * r3r - be * r1i;
        Ai[3][c] = al * r3i + be * r1r;
      }
    }
  }
  for (int r = 0; r < 4; ++r)
    for (int c = 0; c < 4; ++c) {
      mats[p * 32 + (r * 4 + c) * 2 + 0] = Ar[r][c];
      mats[p * 32 + (r * 4 + c) * 2 + 1] = Ai[r][c];
    }
  // B fragments: M[o][k] 8x8 real embedding; B[k][n] = M[n][k], cols n>=8 = 0.
  // B lane layout (4x16, f32): V0 lanes0-15=K0, lanes16-31=K1;
  //                            V1 lanes0-15=K2, lanes16-31=K3.
  for (int ch = 0; ch < 2; ++ch)
    for (int v = 0; v < 2; ++v)
      for (int lane = 0; lane < 32; ++lane) {
        int n = lane & 15;
        int k = ch * 4 + v * 2 + (lane >= 16 ? 1 : 0);
        float val = 0.f;
        if (n < 8) {
          int m = n >> 1, oi = n & 1, c = k >> 1, ki = k & 1;
          val = (oi == 0) ? (ki == 0 ? Ar[m][c] : -Ai[m][c])
                          : (ki == 0 ? Ai[m][c] :  Ar[m][c]);
        }
        bfrag[((p * 2 + ch) * 2 + v) * 32 + lane] = val;
      }
}

// --------------------- pair application (shared code) ----------------------
// Group->LDS element address. FLAT: elements at eb + m*s within 4096 tile.
// ROW: tile is [64 rows][64 cols]; groups stride rows, element = row*64+low.

template <bool ROWMODE>
__device__ __forceinline__ int elem_addr(int g, int m, int s) {
  if constexpr (ROWMODE) {
    int low = g & 63, rg = g >> 6;
    int rb = (rg & (s - 1)) + ((rg & ~(s - 1)) << 2);
    return ((rb + m * s) << 6) + low;
  } else {
    int eb = (g & (s - 1)) + ((g & ~(s - 1)) << 2);
    return eb + m * s;
  }
}

#if USE_WMMA
// Apply pair `p` across the 4096-element LDS tile with v_wmma_f32_16x16x4_f32.
// A operand = 16 groups (rows) x 4 real components (K chunk), B = M^T frags.
template <bool ROWMODE>
__device__ __forceinline__ void apply_pair_wmma(
    float* lre, float* lim, const float* __restrict__ bfrag, int p, int s,
    int lane, int wv) {
  v2f B0, B1;
  B0.x = bfrag[((p * 2 + 0) * 2 + 0) * 32 + lane];
  B0.y = bfrag[((p * 2 + 0) * 2 + 1) * 32 + lane];
  B1.x = bfrag[((p * 2 + 1) * 2 + 0) * 32 + lane];
  B1.y = bfrag[((p * 2 + 1) * 2 + 1) * 32 + lane];
  for (int t = 0; t < 8; ++t) {
    const int gb = (wv * 8 + t) * 16;     // 16 groups per wmma batch
    const int g = gb + (lane & 15);
    const int m0 = (lane >= 16) ? 1 : 0;  // A rows = groups; K = components
    const int a0 = elem_addr<ROWMODE>(g, m0, s);      // comps 0..3
    const int a1 = elem_addr<ROWMODE>(g, 2 + m0, s);  // comps 4..7
    v2f A0, A1;
    A0.x = lre[a0]; A0.y = lim[a0];
    A1.x = lre[a1]; A1.y = lim[a1];
    v8f C = {0.f, 0.f, 0.f, 0.f, 0.f, 0.f, 0.f, 0.f};
    C = __builtin_amdgcn_wmma_f32_16x16x4_f32(false, A0, false, B0, (short)0, C, false, false);
    C = __builtin_amdgcn_wmma_f32_16x16x4_f32(false, A1, false, B1, (short)0, C, false, false);
    // C rows = groups (v: 0..7 / +8 for upper lanes), cols = output comps
    const int col = lane & 15;
    if (col < 8) {
      const int m = col >> 1;
      const bool isim = (col & 1) != 0;
      const int ghalf = (lane >= 16) ? 8 : 0;
#pragma unroll
      for (int v = 0; v < 8; ++v) {
        int addr = elem_addr<ROWMODE>(gb + ghalf + v, m, s);
        if (isim) lim[addr] = C[v]; else lre[addr] = C[v];
      }
    }
  }
}
#else
// VALU fallback: 4 groups per thread, dense 4x4 complex matvec.
template <bool ROWMODE>
__device__ __forceinline__ void apply_pair_valu(
    float* lre, float* lim, const float* __restrict__ M, int s, int tid) {
  float Br[4][4], Bi[4][4];
  for (int r = 0; r < 4; ++r)
    for (int c = 0; c < 4; ++c) {
      Br[r][c] = M[(r * 4 + c) * 2];
      Bi[r][c] = M[(r * 4 + c) * 2 + 1];
    }
  for (int k = 0; k < 4; ++k) {
    int g = tid + 256 * k;
    float xr[4], xi[4];
    for (int m = 0; m < 4; ++m) {
      int a = elem_addr<ROWMODE>(g, m, s);
      xr[m] = lre[a]; xi[m] = lim[a];
    }
    for (int m = 0; m < 4; ++m) {
      float yr = 0.f, yi = 0.f;
      for (int c = 0; c < 4; ++c) {
        yr += Br[m][c] * xr[c] - Bi[m][c] * xi[c];
        yi += Br[m][c] * xi[c] + Bi[m][c] * xr[c];
      }
      int a = elem_addr<ROWMODE>(g, m, s);
      lre[a] = yr; lim[a] = yi;
    }
  }
}
#endif

// ------------------------------- pass 1 ------------------------------------
// Contiguous tile of 4096 amplitudes (SoA in LDS), apply pairs 0..5.

__global__ __launch_bounds__(256) void pass1_kernel(
    const float* __restrict__ qr, const float* __restrict__ qi,
    const float* __restrict__ mats, const float* __restrict__ bfrag,
    float* __restrict__ outre, float* __restrict__ outim) {
  __shared__ __align__(16) float lre[4096];
  __shared__ __align__(16) float lim[4096];
  const int tid = threadIdx.x;
  const int lane = tid & 31;
  const int wv = tid >> 5;
  const size_t gbase = (size_t)blockIdx.x * 4096;
  const unsigned lre0 = lds_addr(lre), lim0 = lds_addr(lim);
  (void)mats; (void)bfrag; (void)lane; (void)wv;

  for (int k = 0; k < 4; ++k) {
    unsigned fo = (unsigned)(k * 1024 + tid * 4);
    async_ld_b128(lre0 + fo * 4, qr + gbase + fo);
    async_ld_b128(lim0 + fo * 4, qi + gbase + fo);
  }
  wait_async0();
  __syncthreads();

  for (int p = 0; p < 6; ++p) {
    const int s = 1 << (2 * p);
#if USE_WMMA
    apply_pair_wmma<false>(lre, lim, bfrag, p, s, lane, wv);
#else
    apply_pair_valu<false>(lre, lim, mats + p * 32, s, tid);
#endif
    __syncthreads();
  }

  for (int k = 0; k < 4; ++k) {
    unsigned fo = (unsigned)(k * 1024 + tid * 4);
    async_st_b128(outre + gbase + fo, lre0 + fo * 4);
    async_st_b128(outim + gbase + fo, lim0 + fo * 4);
  }
  wait_async0();
}

// --------------------------- pass 2 / pass 3 -------------------------------
// 64 (bits 0..5) x 64 (bits ROWSHIFT..ROWSHIFT+5) tile; rows at 2^ROWSHIFT.
// FINAL fuses |psi|^2 reductions + overlap with state0 + probs[:384].

template <int ROWSHIFT, int MIDBITS, int PBASE, bool FINAL>
__global__ __launch_bounds__(256) void pass2_kernel(
    const float* __restrict__ mats, float* __restrict__ sre,
    float* __restrict__ sim, const float* __restrict__ bfrag,
    const float* __restrict__ qr, const float* __restrict__ qi,
    float* __restrict__ partials, float* __restrict__ out) {
  __shared__ __align__(16) float lre[4096];
  __shared__ __align__(16) float lim[4096];
  __shared__ float wred[8][5];
  const int tid = threadIdx.x;
  const int lane = tid & 31;
  const int wv = tid >> 5;
  unsigned b = blockIdx.x;
  unsigned mid = b & ((1u << MIDBITS) - 1);
  unsigned hi = b >> MIDBITS;
  const size_t base = ((size_t)mid << 6) | ((size_t)hi << (ROWSHIFT + 6));
  const unsigned lre0 = lds_addr(lre), lim0 = lds_addr(lim);
  (void)mats;

  for (int i = 0; i < 8; ++i) {
    int r = wv * 8 + i;
    size_t g = base + ((size_t)r << ROWSHIFT) + (unsigned)(lane * 2);
    unsigned lo = (unsigned)((r * 64 + lane * 2) * 4);
    async_ld_b64(lre0 + lo, sre + g);
    async_ld_b64(lim0 + lo, sim + g);
  }
  wait_async0();
  __syncthreads();

  for (int j = 0; j < 3; ++j) {
    const int s = 1 << (2 * j);
#if USE_WMMA
    apply_pair_wmma<true>(lre, lim, bfrag, PBASE + j, s, lane, wv);
#else
    apply_pair_valu<true>(lre, lim, mats + (PBASE + j) * 32, s, tid);
#endif
    __syncthreads();
  }

  if constexpr (FINAL) {
    float sp = 0.f, ss = 0.f, en = 0.f, ovr = 0.f, ovi = 0.f;
    for (int i = 0; i < 8; ++i) {
      int r = wv * 8 + i;
      size_t rowbase = base + ((size_t)r << ROWSHIFT);
#pragma unroll
      for (int j = 0; j < 2; ++j) {
        int li = r * 64 + lane * 2 + j;
        size_t idx = rowbase + (unsigned)(lane * 2 + j);
        float fr = lre[li], fi = lim[li];
        float pr = fr * fr + fi * fi;
        sp += pr;
        ss += ((idx >> 23) & 1) ? -pr : pr;
        float pe = pr + 1e-12f;
        en -= pe * log2f(pe);
        float ar = qr[idx], ai = qi[idx];
        ovr += ar * fr + ai * fi;   // Re(conj(s0)*f)
        ovi += ar * fi - ai * fr;   // Im(conj(s0)*f)
        if (idx < 384u) out[idx] = pr;  // probs_raw[:384]
      }
    }
    float v0 = sp, v1 = ss, v2 = en, v3 = ovr, v4 = ovi;
    for (int o = 16; o > 0; o >>= 1) {
      v0 += __shfl_xor(v0, o); v1 += __shfl_xor(v1, o);
      v2 += __shfl_xor(v2, o); v3 += __shfl_xor(v3, o);
      v4 += __shfl_xor(v4, o);
    }
    if (lane == 0) {
      wred[wv][0] = v0; wred[wv][1] = v1; wred[wv][2] = v2;
      wred[wv][3] = v3; wred[wv][4] = v4;
    }
    __syncthreads();
    if (tid < 5) {
      float s = 0.f;
      for (int w = 0; w < 8; ++w) s += wred[w][tid];
      partials[(size_t)blockIdx.x * 8 + tid] = s;
    }
  } else {
    for (int i = 0; i < 8; ++i) {
      int r = wv * 8 + i;
      size_t g = base + ((size_t)r << ROWSHIFT) + (unsigned)(lane * 2);
      unsigned lo = (unsigned)((r * 64 + lane * 2) * 4);
      async_st_b64(sre + g, lre0 + lo);
      async_st_b64(sim + g, lim0 + lo);
    }
    wait_async0();
  }
}

// ------------------------------ finalize -----------------------------------

__global__ void fin_kernel(const float* __restrict__ partials,
                           float* __restrict__ out) {
  int lane = threadIdx.x & 31;
  float tot[5];
  for (int q = 0; q < 5; ++q) {
    float v = 0.f;
    for (int i = lane; i < 4096; i += 32) v += partials[i * 8 + q];
    for (int o = 16; o > 0; o >>= 1) v += __shfl_xor(v, o);
    tot[q] = v;
  }
  if (lane == 0) {
    float sp = tot[0], ss = tot[1], en = tot[2], ovr = tot[3], ovi = tot[4];
    out[384] = sqrtf(sp);             // state_norm
    out[385] = en;                    // entropy
    out[386] = ovr * ovr + ovi * ovi; // fidelity
    out[387] = ss / sp;               // z_exp
    out[388] = sp / sp;               // x_exp (== 1)
    out[389] = sp / sp;               // y_exp (== 1)
    out[390] = 16.0f;                 // qv = min(dim, depth)^2
  }
}

// ------------------------------- launcher ----------------------------------

extern "C" void kernel_launch(void* const* d_in, const int* in_sizes, int n_in,
                              void* d_out, int out_size, void* d_ws,
                              size_t ws_size, hipStream_t stream) {
  (void)in_sizes; (void)n_in; (void)out_size; (void)ws_size;
  const float* qr  = (const float*)d_in[0];   // qs_real  [2^24]
  const float* qi  = (const float*)d_in[1];   // qs_imag  [2^24]
  const float* rot = (const float*)d_in[2];   // rot_params [4][24][3]
  const float* ew  = (const float*)d_in[3];   // ent_weights [3][12]
  float* out = (float*)d_out;                 // 391 floats

  float* wsf      = (float*)d_ws;
  float* mats     = wsf + 16;    // 12 * 32 floats
  float* bfrag    = wsf + 512;   // 12 * 2 * 2 * 32 floats (WMMA B fragments)
  float* partials = wsf + 4096;  // 4096 blocks * 8 floats
  float* sre = (float*)((char*)d_ws + (size_t)(1u << 20));  // 64MB plane
  float* sim = sre + (size_t)DIM;                           // 64MB plane

  prep_kernel<<<1, 16, 0, stream>>>(rot, ew, mats, bfrag);
  pass1_kernel<<<4096, 256, 0, stream>>>(qr, qi, mats, bfrag, sre, sim);
  pass2_kernel<12, 6, 6, false><<<4096, 256, 0, stream>>>(
      mats, sre, sim, bfrag, nullptr, nullptr, nullptr, nullptr);
  pass2_kernel<18, 12, 9, true><<<4096, 256, 0, stream>>>(
      mats, sre, sim, bfrag, qr, qi, partials, out);
  fin_kernel<<<1, 32, 0, stream>>>(partials, out);
}